// NeuralstackOnly_37512244363490
// MI455X (gfx1250) — compile-verified
//
#include <hip/hip_runtime.h>
#include <cstdint>
#include <math.h>

#define N_STACK 32
#define D_DIM   256
#define ZO      1e-6f
#define EPSF    1e-8f
#define POW_EPSF 1e-12f

// wave32 xor-tree sum using ds_swizzle bitmask mode (no index VGPRs, no clamp
// guards). offset = xor_mask<<10 | or_mask<<5 | and_mask; and=0x1f keeps the
// full 32-lane group (ISA 11.2: SWAPX16/8/4/2/1).
__device__ __forceinline__ float wave_sum32(float v) {
  v += __int_as_float(__builtin_amdgcn_ds_swizzle(__float_as_int(v), 0x401F)); // xor 16
  v += __int_as_float(__builtin_amdgcn_ds_swizzle(__float_as_int(v), 0x201F)); // xor 8
  v += __int_as_float(__builtin_amdgcn_ds_swizzle(__float_as_int(v), 0x101F)); // xor 4
  v += __int_as_float(__builtin_amdgcn_ds_swizzle(__float_as_int(v), 0x081F)); // xor 2
  v += __int_as_float(__builtin_amdgcn_ds_swizzle(__float_as_int(v), 0x041F)); // xor 1
  return v;
}

__device__ __forceinline__ float eluf(float x) {
  return x > 0.0f ? x : expm1f(x);
}

// x is strictly positive and bounded here (>= POW_EPS, <= ~1.7), so the
// hardware log/exp transcendentals are safe: x**s = exp2(s * log2(x)).
__device__ __forceinline__ float pow_pos(float x, float s) {
  return exp2f(s * log2f(x));
}

// One workgroup per batch element. 256 threads = 8 wave32.
// Thread d owns stack[0..31][d] in VGPRs (32 KB of stack state per block,
// entirely register-resident). Pointer (32 floats) + gate scalars in LDS.
// x rows are double-buffered into LDS with CDNA5 async global->LDS loads
// (ASYNCcnt / s_wait_asynccnt path).
__global__ __launch_bounds__(256)
void neural_stack_kernel(const float* __restrict__ x,
                         const float* __restrict__ should_pop,
                         const float* __restrict__ sharpen_in,
                         const float* __restrict__ latch_enable,
                         const float* __restrict__ latch_init,
                         float* __restrict__ out, int T) {
  const int b    = blockIdx.x;
  const int d    = threadIdx.x;      // 0..255 == D index
  const int lane = d & 31;
  const int wave = d >> 5;

  __shared__ float xsh[2][D_DIM];      // async double buffer for x rows
  __shared__ float pbuf[2][N_STACK];   // sharpened pointer, double buffered
  __shared__ float red[8 * 4];         // per-wave reduction partials
  __shared__ float scal[4];            // pop, push, g
  __shared__ float cnorm[2];           // ||should_pop||, ||latch_enable||

  const float sp      = should_pop[d];
  const float le      = latch_enable[d];
  const float sharpen = sharpen_in[0];
  float latch = latch_init[(size_t)b * D_DIM + d];

  float st[N_STACK];
#pragma unroll
  for (int n = 0; n < N_STACK; ++n) st[n] = ZO;

  // ---- loop-invariant norms of the two parameter vectors ----
  {
    float a = wave_sum32(sp * sp);
    float c = wave_sum32(le * le);
    if (lane == 0) { red[wave * 4 + 0] = a; red[wave * 4 + 1] = c; }
    if (d < N_STACK) pbuf[0][d] = (d == 0) ? 1.0f : ZO;  // ptr0
    __syncthreads();
    if (d == 0) {
      float s0 = 0.f, s1 = 0.f;
#pragma unroll
      for (int w = 0; w < 8; ++w) { s0 += red[w * 4 + 0]; s1 += red[w * 4 + 1]; }
      cnorm[0] = fmaxf(sqrtf(s0), EPSF);
      cnorm[1] = fmaxf(sqrtf(s1), EPSF);
    }
    __syncthreads();
  }
  const float an_sp = cnorm[0];
  const float an_le = cnorm[1];

  const float* xb = x   + (size_t)b * T * D_DIM;
  float*       ob = out + (size_t)b * T * D_DIM;

  // ---- prime the async pipeline: x row 0 -> xsh[0] (each lane loads the
  // exact element it will read, so no extra barrier is needed for xsh) ----
  {
    unsigned lds = (unsigned)(uintptr_t)(&xsh[0][d]);
    unsigned go  = (unsigned)(d * 4);
    asm volatile("global_load_async_to_lds_b32 %0, %1, %2"
                 :: "v"(lds), "v"(go), "s"(xb) : "memory");
  }

  for (int t = 0; t < T; ++t) {
    const int pa  = t & 1;       // pointer/x buffer holding state for step t
    const int pnb = pa ^ 1;      // buffer receiving new pointer / next x row

    // wait for x[t] to land in LDS (ASYNCcnt), read own element
    asm volatile("s_wait_asynccnt 0" ::: "memory");
    const float inp = xsh[pa][d];

    // kick off async prefetch of x[t+1] into the other buffer; it has the
    // whole step's compute (reductions + 3 barriers) to cover its latency
    if (t + 1 < T) {
      unsigned lds = (unsigned)(uintptr_t)(&xsh[pnb][d]);
      unsigned go  = (unsigned)(d * 4);
      const float* xn = xb + (size_t)(t + 1) * D_DIM;
      asm volatile("global_load_async_to_lds_b32 %0, %1, %2"
                   :: "v"(lds), "v"(go), "s"(xn) : "memory");
    }

    // ---- 4 reductions over D: sp.latch, |latch|^2, le.inp, |inp|^2 ----
    float v0 = wave_sum32(sp * latch);
    float v1 = wave_sum32(latch * latch);
    float v2 = wave_sum32(le * inp);
    float v3 = wave_sum32(inp * inp);
    if (lane == 0) {
      red[wave * 4 + 0] = v0; red[wave * 4 + 1] = v1;
      red[wave * 4 + 2] = v2; red[wave * 4 + 3] = v3;
    }
    __syncthreads();                       // B1
    if (d == 0) {
      float dt = 0.f, nl = 0.f;
#pragma unroll
      for (int w = 0; w < 8; ++w) { dt += red[w * 4 + 0]; nl += red[w * 4 + 1]; }
      float pop = eluf(dt / (an_sp * fmaxf(sqrtf(nl), EPSF)));
      scal[0] = pop;
      scal[1] = 1.0f - pop;
    } else if (d == 1) {
      float dt = 0.f, ni = 0.f;
#pragma unroll
      for (int w = 0; w < 8; ++w) { dt += red[w * 4 + 2]; ni += red[w * 4 + 3]; }
      scal[2] = eluf(dt / (an_le * fmaxf(sqrtf(ni), EPSF)));
    }
    __syncthreads();                       // B2
    const float pop  = scal[0];
    const float push = scal[1];
    const float g    = scal[2];

    // ---- pointer update + sharpen on wave 0 (lane n == stack slot n);
    // runs concurrently with the stack update on all 8 waves ----
    if (d < N_STACK) {
      const int n = d;
      const float ppush = pbuf[pa][(n + N_STACK - 1) & (N_STACK - 1)];
      const float ppop  = pbuf[pa][(n + 1) & (N_STACK - 1)];
      const float np    = push * ppush + pop * ppop;
      const float pw    = pow_pos(fmaxf(np, POW_EPSF), sharpen); // relu folds into max
      const float ssum  = wave_sum32(pw);
      pbuf[pnb][n] = pw / fmaxf(ssum, EPSF);
    }

    // ---- stack update with OLD pointer (all threads; register resident) ----
#pragma unroll
    for (int n = 0; n < N_STACK; ++n) {
      const float ppush = pbuf[pa][(n + N_STACK - 1) & (N_STACK - 1)];
      const float pold  = pbuf[pa][n];
      const float stp = st[n] * (1.0f - ppush) + inp * ppush;   // push branch
      const float sto = st[n] * (1.0f - pold)  + ZO  * pold;    // pop branch
      st[n] = push * stp + pop * sto;
    }
    __syncthreads();                       // B3: new sharpened pointer visible

    // ---- read with NEW pointer, gate, emit ----
    float s = 0.f;
#pragma unroll
    for (int n = 0; n < N_STACK; ++n)
      s += st[n] * pbuf[pnb][n];
    ob[(size_t)t * D_DIM + d] = pop * s;

    // ---- latch EMA (uses inp-gate g) ----
    latch = g * inp + (1.0f - g) * latch;
  }
}

extern "C" void kernel_launch(void* const* d_in, const int* in_sizes, int n_in,
                              void* d_out, int out_size, void* d_ws, size_t ws_size,
                              hipStream_t stream) {
  const float* x            = (const float*)d_in[0];
  const float* should_pop   = (const float*)d_in[1];
  const float* sharpen      = (const float*)d_in[2];
  const float* latch_enable = (const float*)d_in[3];
  const float* latch_init   = (const float*)d_in[4];
  float* out = (float*)d_out;

  const int D = in_sizes[1];                 // 256 (kernel assumes this)
  if (D != D_DIM) return;
  const int B = in_sizes[4] / D;             // 128
  const int T = in_sizes[0] / (B * D);       // 256

  neural_stack_kernel<<<B, 256, 0, stream>>>(x, should_pop, sharpen,
                                             latch_enable, latch_init, out, T);
}